// VectorGauntTensorProductAllParitiesS2Grid_56401510531327
// MI455X (gfx1250) — compile-verified
//
#include <hip/hip_runtime.h>
#include <hip/hip_bf16.h>

typedef __attribute__((ext_vector_type(2))) float v2f;
typedef __attribute__((ext_vector_type(8))) float v8f;

#define NN   512
#define FIN  32
#define CC   32
#define KK   25
#define KP   28          // K padded to 7 chunks of 4 for wmma inner dim
#define VV   3
#define GGR  2016
#define GT   16
#define NGT  (GGR / GT)  // 126
#define NB   2           // nodes per workgroup
#define ROWS (NB * CC)   // 64 (n,c) rows per component
#define NWAVE (ROWS / 16)
#define NTHR (NWAVE * 32) // 128
#define KS   34          // H row stride (bank-conflict-free, 8B aligned pairs)
#define YTS  30          // transposed-Y tile row stride
#define YWS  18          // Yw tile row stride
#define SCRS 18          // cr scratch row stride

__device__ __forceinline__ int l_of(int k) {
    return (k < 1) ? 0 : (k < 4) ? 1 : (k < 9) ? 2 : (k < 16) ? 3 : 4;
}

__device__ __forceinline__ v8f wmma_f32(v2f a, v2f b, v8f c) {
    return __builtin_amdgcn_wmma_f32_16x16x4_f32(false, a, false, b, (short)0, c,
                                                 false, false);
}

__global__ __launch_bounds__(NTHR)
void gaunt_s2_kernel(const float* __restrict__ x1, const float* __restrict__ x2,
                     const float* __restrict__ W1s, const float* __restrict__ W2s,
                     const float* __restrict__ Wouts, const float* __restrict__ Y,
                     const float* __restrict__ qw, float* __restrict__ out)
{
    __shared__ __align__(16) float H1[VV * ROWS * KS];    // h1, reused as CO at end
    __shared__ __align__(16) float H2[VV * ROWS * KS];
    __shared__ __align__(16) float YT[2][GT * YTS];       // Y tile, [g][k] (B frags, fwd)
    __shared__ __align__(16) float YW[2][32 * YWS];       // Y*qw tile, [k][g] (B frags, inv)
    __shared__ __align__(16) float SCR[NWAVE * 16 * SCRS]; // per-wave cr transpose scratch

    const int p    = blockIdx.y;
    const int n0   = blockIdx.x * NB;
    const int t    = threadIdx.x;
    const int w    = t >> 5;
    const int L    = t & 31;
    const int col  = L & 15;
    const int half = L >> 4;
    const int r0   = w * 16;

    const float* W1   = W1s   + p * 5 * FIN * CC;
    const float* W2   = W2s   + p * 5 * FIN * CC;
    const float* Wout = Wouts + p * 5 * CC * CC;

    // per-thread slice of the 32x16 Y tile: 4 elements (512 / 128 threads)
    const int yj = t & 15;          // g within tile
    const int yk0 = t >> 4;         // k row base (stride 8 over i)

    // ---- prologue: fetch tile 0 into registers ----
    float yreg[4], qreg[4];
    #pragma unroll
    for (int i = 0; i < 4; ++i) {
        int k = yk0 + i * 8;
        yreg[i] = (k < KK) ? Y[k * GGR + yj] : 0.f;
        qreg[i] = qw[yj];
    }

    // ---- Stage 1: per-degree equivariant channel mix: H[v][row=(n,c)][k] ----
    for (int idx = t; idx < ROWS * VV * KP; idx += NTHR) {
        int k   = idx % KP;
        int v   = (idx / KP) % VV;
        int row = idx / (KP * VV);
        int nl = row >> 5, c = row & 31;
        float a1 = 0.f, a2 = 0.f;
        if (k < KK) {
            int l = l_of(k);
            const float* w1  = W1 + (l * FIN) * CC + c;
            const float* w2  = W2 + (l * FIN) * CC + c;
            const float* xa  = x1 + (((size_t)(n0 + nl) * FIN) * VV + v) * KK + k;
            const float* xb  = x2 + (((size_t)(n0 + nl) * FIN) * VV + v) * KK + k;
            #pragma unroll 4
            for (int f = 0; f < FIN; ++f) {
                a1 += xa[f * VV * KK] * w1[f * CC];
                a2 += xb[f * VV * KK] * w2[f * CC];
            }
        }
        H1[(v * ROWS + row) * KS + k] = a1;
        H2[(v * ROWS + row) * KS + k] = a2;
    }

    // ---- write tile 0 to LDS buffer 0 ----
    #pragma unroll
    for (int i = 0; i < 4; ++i) {
        int k = yk0 + i * 8;
        if (k < KP) YT[0][yj * YTS + k] = yreg[i];
        YW[0][k * YWS + yj] = yreg[i] * qreg[i];
    }

    // co accumulators: [v][k-tile] in WMMA C/D layout (rows=(n,c), cols=k)
    v8f co[VV][2];
    #pragma unroll
    for (int v = 0; v < VV; ++v) { co[v][0] = 0.f; co[v][1] = 0.f; }

    for (int gt = 0; gt < NGT; ++gt) {
        const int cur = gt & 1;
        const bool more = (gt + 1) < NGT;
        // single barrier per iteration: publishes buf[cur] stores (prev iter)
        // and retires all reads of buf[1-cur] before we overwrite it below.
        __syncthreads();

        // issue next tile's global loads early; latency hides under the WMMAs
        if (more) {
            const int gn = (gt + 1) * GT;
            #pragma unroll
            for (int i = 0; i < 4; ++i) {
                int k = yk0 + i * 8;
                yreg[i] = (k < KK) ? Y[k * GGR + gn + yj] : 0.f;
                qreg[i] = qw[gn + yj];
            }
        }

        const float* YTc = YT[cur];
        const float* YWc = YW[cur];

        // ---- forward SH->grid: 6 tiles (2 signals x 3 components), inner K=28 ----
        v8f d[6];
        #pragma unroll
        for (int m = 0; m < 6; ++m) d[m] = 0.f;
        #pragma unroll
        for (int kk = 0; kk < 7; ++kk) {
            const int k0 = kk * 4 + half * 2;
            v2f b = *(const v2f*)&YTc[col * YTS + k0];
            #pragma unroll
            for (int v = 0; v < VV; ++v) {
                const int hoff = (v * ROWS + r0 + col) * KS + k0;
                v2f a1 = *(const v2f*)&H1[hoff];
                v2f a2 = *(const v2f*)&H2[hoff];
                d[v]     = wmma_f32(a1, b, d[v]);
                d[3 + v] = wmma_f32(a2, b, d[3 + v]);
            }
        }

        // ---- pointwise vector cross product on the sphere (same lane slots) ----
        v8f cr[3];
        cr[0] = d[1] * d[5] - d[2] * d[4];
        cr[1] = d[2] * d[3] - d[0] * d[5];
        cr[2] = d[0] * d[4] - d[1] * d[3];

        // ---- inverse grid->SH (quadrature): co += cr * Yw^T ----
        float* scr = &SCR[w * 16 * SCRS];
        #pragma unroll
        for (int v = 0; v < VV; ++v) {
            // C/D layout -> A layout via per-wave LDS scratch (in-order DS, no barrier)
            #pragma unroll
            for (int j = 0; j < 8; ++j)
                scr[(j + half * 8) * SCRS + col] = cr[v][j];
            #pragma unroll
            for (int gg = 0; gg < 4; ++gg) {
                const int goff = gg * 4 + half * 2;
                v2f a  = *(const v2f*)&scr[col * SCRS + goff];
                v2f b0 = *(const v2f*)&YWc[col * YWS + goff];
                v2f b1 = *(const v2f*)&YWc[(16 + col) * YWS + goff];
                co[v][0] = wmma_f32(a, b0, co[v][0]);
                co[v][1] = wmma_f32(a, b1, co[v][1]);
            }
        }

        // ---- park next tile into the alternate LDS buffer ----
        if (more) {
            const int nxt = 1 - cur;
            #pragma unroll
            for (int i = 0; i < 4; ++i) {
                int k = yk0 + i * 8;
                if (k < KP) YT[nxt][yj * YTS + k] = yreg[i];
                YW[nxt][k * YWS + yj] = yreg[i] * qreg[i];
            }
        }
    }

    // ---- write co accumulators to LDS (reuse H1) ----
    __syncthreads();
    float* CO = H1;
    #pragma unroll
    for (int v = 0; v < VV; ++v)
        #pragma unroll
        for (int kt = 0; kt < 2; ++kt)
            #pragma unroll
            for (int j = 0; j < 8; ++j)
                CO[(v * ROWS + r0 + j + half * 8) * KS + kt * 16 + col] = co[v][kt][j];
    __syncthreads();

    // ---- Stage 5: output equivariant mix + store ----
    for (int idx = t; idx < NB * CC * VV * KK; idx += NTHR) {
        int k  = idx % KK;
        int v  = (idx / KK) % VV;
        int b  = (idx / (KK * VV)) % CC;
        int nl = idx / (KK * VV * CC);
        int l  = l_of(k);
        const float* wo = Wout + (l * CC) * CC + b;
        float acc = 0.f;
        #pragma unroll 4
        for (int a = 0; a < CC; ++a)
            acc += CO[(v * ROWS + nl * 32 + a) * KS + k] * wo[a * CC];
        out[(((size_t)(n0 + nl) * (3 * CC) + p * CC + b) * VV + v) * KK + k] = acc;
    }
}

extern "C" void kernel_launch(void* const* d_in, const int* in_sizes, int n_in,
                              void* d_out, int out_size, void* d_ws, size_t ws_size,
                              hipStream_t stream) {
    (void)in_sizes; (void)n_in; (void)d_ws; (void)ws_size; (void)out_size;
    const float* x1    = (const float*)d_in[0];
    const float* x2    = (const float*)d_in[1];
    const float* W1s   = (const float*)d_in[2];
    const float* W2s   = (const float*)d_in[3];
    const float* Wouts = (const float*)d_in[4];
    const float* Y     = (const float*)d_in[5];
    const float* qw    = (const float*)d_in[6];
    float* out = (float*)d_out;

    dim3 grid(NN / NB, 3);
    dim3 block(NTHR);
    gaunt_s2_kernel<<<grid, block, 0, stream>>>(x1, x2, W1s, W2s, Wouts, Y, qw, out);
}